// GATConv_17008070492255
// MI455X (gfx1250) — compile-verified
//
#include <hip/hip_runtime.h>
#include <hip/hip_bf16.h>
#include <math.h>

#define D_IN 64
#define H    4
#define F    16
#define HF   64           // H*F == D_IN here
#define EFST 1e-3f

typedef __attribute__((ext_vector_type(2))) float v2f;
typedef __attribute__((ext_vector_type(8))) float v8f;

// ---- order-preserving float<->uint encoding for atomic max over signed floats
__device__ __forceinline__ unsigned enc_f32(float x) {
    unsigned u = __float_as_uint(x);
    return (x >= 0.0f) ? (u | 0x80000000u) : ~u;
}
__device__ __forceinline__ float dec_f32(unsigned u) {
    return (u & 0x80000000u) ? __uint_as_float(u & 0x7FFFFFFFu)
                             : __uint_as_float(~u);
}
__device__ __forceinline__ float dot4(float4 a, float4 b) {
    return fmaf(a.x, b.x, fmaf(a.y, b.y, fmaf(a.z, b.z, a.w * b.w)));
}

// encoded(-inf): asuint(-inf)=0xFF800000 -> ~ = 0x007FFFFF
#define ENC_NEG_INF 0x007FFFFFu

// ---------------------------------------------------------------- init
__global__ void k_init(float* __restrict__ out, float* __restrict__ denom,
                       unsigned* __restrict__ menc, int n_nodes) {
    int i = blockIdx.x * blockDim.x + threadIdx.x;
    if (i < n_nodes * HF) out[i] = 0.0f;
    if (i < n_nodes * H) { denom[i] = 0.0f; menc[i] = ENC_NEG_INF; }
}

// ------------------------------------------------- z = feat @ W^T via WMMA f32
// One wave per 16x16 output tile. K=64 -> 16 x V_WMMA_F32_16X16X4_F32.
// fp32 A 16x4 layout (ISA 7.12.2): lanes 0-15 hold K=0,1; lanes 16-31 hold K=2,3.
// B 4x16 mirrors (K halves across lane halves). C/D: VGPR v, lane half*16+n ->
// row M = v + 8*half, col N = n.
__global__ void __launch_bounds__(256)
k_gemm_wmma(const float* __restrict__ feat, const float* __restrict__ W,
            float* __restrict__ z, int n_nodes) {
    const int waves_per_block = 256 / 32;
    int wave = blockIdx.x * waves_per_block + (threadIdx.x >> 5);
    int lane = threadIdx.x & 31;

    const int colTiles = HF / 16;             // 4
    int rowBlock = wave / colTiles;
    int colBlock = wave - rowBlock * colTiles;
    int row0 = rowBlock * 16;
    if (row0 >= n_nodes) return;              // whole-wave uniform exit
    int col0 = colBlock * 16;

    int half = lane >> 4;                     // 0|1 : which K-pair half
    int l    = lane & 15;                     // row (A) / col (B) index

    int ra = row0 + l; if (ra > n_nodes - 1) ra = n_nodes - 1;  // tail clamp
    const float* arow = feat + (size_t)ra * D_IN + half * 2;
    const float* brow = W    + (size_t)(col0 + l) * D_IN + half * 2;

    v8f c = {};
#pragma unroll
    for (int kk = 0; kk < D_IN / 4; ++kk) {
        v2f a = *(const v2f*)(arow + kk * 4);
        v2f b = *(const v2f*)(brow + kk * 4);
        // (neg_a, A, neg_b, B, c_mod, C, reuse_a, reuse_b)
        c = __builtin_amdgcn_wmma_f32_16x16x4_f32(false, a, false, b,
                                                  (short)0, c, false, false);
    }
#pragma unroll
    for (int v = 0; v < 8; ++v) {
        int r = row0 + half * 8 + v;
        if (r < n_nodes) z[(size_t)r * HF + col0 + l] = c[v];
    }
}

// ----------------------------------------- per-node: el, er, inv_s (clamped)
__global__ void k_node(const float* __restrict__ z,
                       const float* __restrict__ attn_l,
                       const float* __restrict__ attn_r,
                       float* __restrict__ el, float* __restrict__ er,
                       float* __restrict__ inv_s, int n_nodes) {
    int n = blockIdx.x * blockDim.x + threadIdx.x;
    if (n >= n_nodes) return;
    const float4* zr = (const float4*)(z + (size_t)n * HF);
    const float4* al4 = (const float4*)attn_l;
    const float4* ar4 = (const float4*)attn_r;
    float s[F];
#pragma unroll
    for (int f = 0; f < F; ++f) s[f] = 0.0f;
#pragma unroll
    for (int h = 0; h < H; ++h) {
        float elh = 0.0f, erh = 0.0f;
#pragma unroll
        for (int q = 0; q < 4; ++q) {
            float4 zv = zr[h * 4 + q];
            elh += dot4(zv, al4[h * 4 + q]);
            erh += dot4(zv, ar4[h * 4 + q]);
            s[q * 4 + 0] = fmaf(zv.x, zv.x, s[q * 4 + 0]);
            s[q * 4 + 1] = fmaf(zv.y, zv.y, s[q * 4 + 1]);
            s[q * 4 + 2] = fmaf(zv.z, zv.z, s[q * 4 + 2]);
            s[q * 4 + 3] = fmaf(zv.w, zv.w, s[q * 4 + 3]);
        }
        el[n * H + h] = elh;
        er[n * H + h] = erh;
    }
#pragma unroll
    for (int f = 0; f < F; ++f)
        inv_s[(size_t)n * F + f] = 1.0f / fmaxf(s[f], EFST);
}

// ------------------------------------- per-edge: score + atomic segment max
__global__ void k_edge_score(const float* __restrict__ z,
                             const int* __restrict__ src,
                             const int* __restrict__ dst,
                             const float* __restrict__ el,
                             const float* __restrict__ er,
                             const float* __restrict__ inv_s,
                             float* __restrict__ escore,
                             unsigned* __restrict__ menc, int n_edges) {
    int e = blockIdx.x * blockDim.x + threadIdx.x;
    if (e >= n_edges) return;
    int u = src[e], v = dst[e];
    const float4* zu = (const float4*)(z + (size_t)u * HF);
    const float4* zv = (const float4*)(z + (size_t)v * HF);
    float edp[H];
#pragma unroll
    for (int h = 0; h < H; ++h) {
        float acc = 0.0f;
#pragma unroll
        for (int q = 0; q < 4; ++q) acc += dot4(zu[h * 4 + q], zv[h * 4 + q]);
        edp[h] = acc;
    }
    const float4* iu = (const float4*)(inv_s + (size_t)u * F);
    const float4* iv = (const float4*)(inv_s + (size_t)v * F);
    float ss = 0.0f;
#pragma unroll
    for (int q = 0; q < 4; ++q) ss += dot4(iu[q], iv[q]);
#pragma unroll
    for (int h = 0; h < H; ++h) {
        float t = edp[h] * ss;
        t = t > 0.0f ? t : 0.0f;                       // relu
        float es = t * (el[u * H + h] + er[v * H + h]);
        escore[(size_t)e * H + h] = es;
        atomicMax(&menc[(size_t)v * H + h], enc_f32(es));
    }
}

// -------------------- per-(edge,head): exp(es - m[dst]) in place + denom sum
__global__ void k_exp(const int* __restrict__ dst,
                      const unsigned* __restrict__ menc,
                      float* __restrict__ escore,   // in: score, out: ex
                      float* __restrict__ denom, int n_edges) {
    int i = blockIdx.x * blockDim.x + threadIdx.x;
    if (i >= n_edges * H) return;
    int e = i >> 2;           // H == 4
    int h = i & 3;
    int v = dst[e];
    float m = dec_f32(menc[v * H + h]);
    float ex = __expf(escore[i] - m);
    escore[i] = ex;
    atomicAdd(&denom[v * H + h], ex);
}

// ------------- per-(edge,feature): out[dst] += (ex/denom) * z[src], atomically
__global__ void k_agg(const int* __restrict__ src, const int* __restrict__ dst,
                      const float* __restrict__ z,
                      const float* __restrict__ ex,
                      const float* __restrict__ denom,
                      float* __restrict__ out, int n_edges) {
    long long i = (long long)blockIdx.x * blockDim.x + threadIdx.x;
    if (i >= (long long)n_edges * HF) return;
    int e = (int)(i >> 6);
    int t = (int)(i & 63);
    int h = t >> 4;
    int u = src[e], v = dst[e];
    float d = denom[v * H + h];
    float a = ex[(size_t)e * H + h] / (d > 0.0f ? d : 1.0f);
    atomicAdd(&out[(size_t)v * HF + t], a * z[(size_t)u * HF + t]);
}

extern "C" void kernel_launch(void* const* d_in, const int* in_sizes, int n_in,
                              void* d_out, int out_size, void* d_ws, size_t ws_size,
                              hipStream_t stream) {
    const float* feat   = (const float*)d_in[0];
    const int*   src    = (const int*)d_in[1];
    const int*   dst    = (const int*)d_in[2];
    const float* W      = (const float*)d_in[3];
    const float* attn_l = (const float*)d_in[4];
    const float* attn_r = (const float*)d_in[5];
    float*       out    = (float*)d_out;

    const int n_nodes = in_sizes[0] / D_IN;
    const int n_edges = in_sizes[1];

    // workspace layout (floats):  z | el | er | inv_s | denom | menc | escore
    float*    ws     = (float*)d_ws;
    float*    z      = ws;                                   // n*64
    float*    el     = z + (size_t)n_nodes * HF;             // n*4
    float*    er     = el + (size_t)n_nodes * H;             // n*4
    float*    inv_s  = er + (size_t)n_nodes * H;             // n*16
    float*    denom  = inv_s + (size_t)n_nodes * F;          // n*4
    unsigned* menc   = (unsigned*)(denom + (size_t)n_nodes * H); // n*4
    float*    escore = (float*)(menc + (size_t)n_nodes * H); // e*4 (reused as ex)

    const int TPB = 256;

    // 1) init out/denom/max
    {
        int total = n_nodes * HF;
        k_init<<<(total + TPB - 1) / TPB, TPB, 0, stream>>>(out, denom, menc, n_nodes);
    }
    // 2) z = feat @ W^T  (WMMA f32 16x16x4, one wave per 16x16 tile)
    {
        int rowBlocks = (n_nodes + 15) / 16;
        int tiles = rowBlocks * (HF / 16);
        int wavesPerBlock = TPB / 32;
        int blocks = (tiles + wavesPerBlock - 1) / wavesPerBlock;
        k_gemm_wmma<<<blocks, TPB, 0, stream>>>(feat, W, z, n_nodes);
    }
    // 3) node-level el/er/inv_s
    k_node<<<(n_nodes + TPB - 1) / TPB, TPB, 0, stream>>>(z, attn_l, attn_r,
                                                          el, er, inv_s, n_nodes);
    // 4) edge scores + segment max
    k_edge_score<<<(n_edges + TPB - 1) / TPB, TPB, 0, stream>>>(
        z, src, dst, el, er, inv_s, escore, menc, n_edges);
    // 5) exp + denom
    {
        int total = n_edges * H;
        k_exp<<<(total + TPB - 1) / TPB, TPB, 0, stream>>>(dst, menc, escore,
                                                           denom, n_edges);
    }
    // 6) weighted aggregation
    {
        long long total = (long long)n_edges * HF;
        int blocks = (int)((total + TPB - 1) / TPB);
        k_agg<<<blocks, TPB, 0, stream>>>(src, dst, z, escore, denom, out, n_edges);
    }
}